// GraphAttention_22273700397226
// MI455X (gfx1250) — compile-verified
//
#include <hip/hip_runtime.h>

typedef __attribute__((ext_vector_type(2))) float v2f;
typedef __attribute__((ext_vector_type(4))) float v4f;
typedef __attribute__((ext_vector_type(8))) float v8f;

#define N_NODES 1024
#define NBATCH  32
#define F       64
#define WAVES   8
#define ROWS_PER_WAVE 16
#define ROWS_PER_WG   (WAVES * ROWS_PER_WAVE)   // 128
#define WGS_PER_BATCH (N_NODES / ROWS_PER_WG)   // 8
#define WSTRIDE 80   // padded LDS row stride for weight tiles (bank-conflict free B frags)
#define KSTRIDE 68   // padded LDS row stride for K/V tiles
#define SSTRIDE 18   // padded LDS row stride for per-wave S scratch
#define KT      32   // k-tile rows streamed per iteration

// D = A(16x4 f32) * B(4x16 f32) + C(16x16 f32), full fp32 on the WMMA pipe.
__device__ __forceinline__ v8f wmma4(v2f a, v2f b, v8f c) {
  return __builtin_amdgcn_wmma_f32_16x16x4_f32(
      /*neg_a=*/false, a, /*neg_b=*/false, b,
      /*c_mod=*/(short)0, c, /*reuse_a=*/false, /*reuse_b=*/false);
}

// ---------------------------------------------------------------------------
// Phase 0: Kb = X@attW, Vb = X@W, Sf = X@selfW + bias     (all [B,N,64] f32)
// ---------------------------------------------------------------------------
__global__ __launch_bounds__(256) void gat_phase0(
    const float* __restrict__ X, const float* __restrict__ W,
    const float* __restrict__ attW, const float* __restrict__ selfW,
    const float* __restrict__ biasW,
    float* __restrict__ Kb, float* __restrict__ Vb, float* __restrict__ Sf)
{
  __shared__ float sWgt[3][F * WSTRIDE];
  __shared__ float sBias[F];

  const int tid  = threadIdx.x;
  const int lane = tid & 31;
  const int wave = tid >> 5;
  const int half = lane >> 4;   // 0: K0/K1 lanes, 1: K2/K3 lanes
  const int ln   = lane & 15;

  const int b    = blockIdx.x / WGS_PER_BATCH;
  const int rc   = blockIdx.x % WGS_PER_BATCH;
  const int row0 = rc * ROWS_PER_WG + wave * ROWS_PER_WAVE;

  // Stage all three weight matrices into padded LDS.
  for (int i = tid; i < F * F; i += 256) {
    int r = i >> 6, c = i & 63;
    sWgt[0][r * WSTRIDE + c] = attW[i];
    sWgt[1][r * WSTRIDE + c] = W[i];
    sWgt[2][r * WSTRIDE + c] = selfW[i];
  }
  if (tid < F) sBias[tid] = biasW[tid];
  __syncthreads();

  // A fragments of this wave's 16 rows of X (register resident, reused 12x).
  v2f xf[16];
  const float* xrow = X + ((size_t)b * N_NODES + row0 + ln) * F;
  #pragma unroll
  for (int t = 0; t < 16; ++t)
    xf[t] = *(const v2f*)(xrow + 4 * t + 2 * half);

  #pragma unroll
  for (int m = 0; m < 3; ++m) {
    float* dst = (m == 0) ? Kb : (m == 1) ? Vb : Sf;
    const float* wm = sWgt[m];
    #pragma unroll
    for (int g = 0; g < 4; ++g) {
      v8f acc = {0.f, 0.f, 0.f, 0.f, 0.f, 0.f, 0.f, 0.f};
      #pragma unroll
      for (int t = 0; t < 16; ++t) {
        int kk = 4 * t + 2 * half;
        v2f bf;
        bf.x = wm[kk * WSTRIDE + g * 16 + ln];
        bf.y = wm[(kk + 1) * WSTRIDE + g * 16 + ln];
        acc = wmma4(xf[t], bf, acc);
      }
      #pragma unroll
      for (int r = 0; r < 8; ++r) {
        int mrow = r + 8 * half;     // C/D layout: VGPR r -> row r (+8 upper half)
        float v = acc[r];
        if (m == 2) v += sBias[g * 16 + ln];
        dst[((size_t)b * N_NODES + row0 + mrow) * F + g * 16 + ln] = v;
      }
    }
  }
}

// ---------------------------------------------------------------------------
// Phase 1: fused attention.  For each 16-row block:
//   S = (Kb_rows * scale) @ Kb_tile^T  -> mask diag, leaky_relu
//   O += S @ Vb_tile ;  H = O + Sf
// ---------------------------------------------------------------------------
__global__ __launch_bounds__(256) void gat_phase1(
    const float* __restrict__ a,
    const float* __restrict__ Kb, const float* __restrict__ Vb,
    const float* __restrict__ Sf, float* __restrict__ H)
{
  __shared__ float sK[KT * KSTRIDE];
  __shared__ float sV[KT * KSTRIDE];
  __shared__ float sS[WAVES][16 * SSTRIDE];
  __shared__ float sScale[F];

  const int tid  = threadIdx.x;
  const int lane = tid & 31;
  const int wave = tid >> 5;
  const int half = lane >> 4;
  const int ln   = lane & 15;

  const int b    = blockIdx.x / WGS_PER_BATCH;
  const int rc   = blockIdx.x % WGS_PER_BATCH;
  const int row0 = rc * ROWS_PER_WG + wave * ROWS_PER_WAVE;

  // scale[j] = sum_i a[i][j]  (tiny; recomputed per WG, stays in L2)
  if (tid < F) {
    float s = 0.f;
    for (int i = 0; i < F; ++i) s += a[i * F + tid];
    sScale[tid] = s;
  }
  __syncthreads();

  // Q fragments: attX rows scaled per-feature, register resident for whole loop.
  v2f qf[16];
  {
    const float* qrow = Kb + ((size_t)b * N_NODES + row0 + ln) * F;
    #pragma unroll
    for (int t = 0; t < 16; ++t) {
      int kk = 4 * t + 2 * half;
      v2f v = *(const v2f*)(qrow + kk);
      v.x *= sScale[kk];
      v.y *= sScale[kk + 1];
      qf[t] = v;
    }
  }

  v8f o0 = {0.f,0.f,0.f,0.f,0.f,0.f,0.f,0.f};
  v8f o1 = o0, o2 = o0, o3 = o0;

  const float* kbase = Kb + (size_t)b * N_NODES * F;
  const float* vbase = Vb + (size_t)b * N_NODES * F;

  for (int kt = 0; kt < N_NODES; kt += KT) {
    // Cooperative stage of K/V tiles (float4, padded rows keep 16B alignment).
    for (int i = tid; i < KT * 16; i += 256) {
      int r = i >> 4, c4 = (i & 15) << 2;
      *(v4f*)&sK[r * KSTRIDE + c4] = *(const v4f*)&kbase[(size_t)(kt + r) * F + c4];
      *(v4f*)&sV[r * KSTRIDE + c4] = *(const v4f*)&vbase[(size_t)(kt + r) * F + c4];
    }
    if (kt + KT < N_NODES) {
      __builtin_prefetch(&kbase[(size_t)(kt + KT) * F], 0, 0);
      __builtin_prefetch(&vbase[(size_t)(kt + KT) * F], 0, 0);
    }
    __syncthreads();

    #pragma unroll
    for (int kb = 0; kb < KT; kb += 16) {
      // ---- S = Q @ Ksub^T  (reduction over 64 features, 16 WMMA) ----
      v8f s = {0.f,0.f,0.f,0.f,0.f,0.f,0.f,0.f};
      #pragma unroll
      for (int t = 0; t < 16; ++t) {
        int kk = 4 * t + 2 * half;
        v2f bf = *(const v2f*)&sK[(kb + ln) * KSTRIDE + kk];  // column ln of Ksub^T
        s = wmma4(qf[t], bf, s);
      }
      // ---- mask diagonal + leaky_relu, reshape D-layout -> A-layout via LDS ----
      int colg = kt + kb + ln;
      #pragma unroll
      for (int r = 0; r < 8; ++r) {
        int mrow = r + 8 * half;
        float v = s[r];
        v = (v > 0.f) ? v : 0.01f * v;
        if (row0 + mrow == colg) v = 0.f;
        sS[wave][mrow * SSTRIDE + ln] = v;
      }
      // ---- O += S @ Vsub  (reduction over 16, 4x4 WMMA) ----
      #pragma unroll
      for (int t = 0; t < 4; ++t) {
        int kk = 4 * t + 2 * half;
        v2f af = *(const v2f*)&sS[wave][ln * SSTRIDE + kk];
        v2f bf;
        bf.x = sV[(kb + kk) * KSTRIDE + 0 * 16 + ln];
        bf.y = sV[(kb + kk + 1) * KSTRIDE + 0 * 16 + ln];
        o0 = wmma4(af, bf, o0);
        bf.x = sV[(kb + kk) * KSTRIDE + 1 * 16 + ln];
        bf.y = sV[(kb + kk + 1) * KSTRIDE + 1 * 16 + ln];
        o1 = wmma4(af, bf, o1);
        bf.x = sV[(kb + kk) * KSTRIDE + 2 * 16 + ln];
        bf.y = sV[(kb + kk + 1) * KSTRIDE + 2 * 16 + ln];
        o2 = wmma4(af, bf, o2);
        bf.x = sV[(kb + kk) * KSTRIDE + 3 * 16 + ln];
        bf.y = sV[(kb + kk + 1) * KSTRIDE + 3 * 16 + ln];
        o3 = wmma4(af, bf, o3);
      }
    }
    __syncthreads();
  }

  // H = O + (X@selfW + bias)
  #pragma unroll
  for (int g = 0; g < 4; ++g) {
    const v8f& o = (g == 0) ? o0 : (g == 1) ? o1 : (g == 2) ? o2 : o3;
    #pragma unroll
    for (int r = 0; r < 8; ++r) {
      int mrow = r + 8 * half;
      size_t idx = ((size_t)b * N_NODES + row0 + mrow) * F + g * 16 + ln;
      H[idx] = o[r] + Sf[idx];
    }
  }
}

// ---------------------------------------------------------------------------
extern "C" void kernel_launch(void* const* d_in, const int* in_sizes, int n_in,
                              void* d_out, int out_size, void* d_ws, size_t ws_size,
                              hipStream_t stream) {
  // setup_inputs order: X, A(unused), W, att_W, a, bias_W, self_W
  const float* X     = (const float*)d_in[0];
  const float* W     = (const float*)d_in[2];
  const float* attW  = (const float*)d_in[3];
  const float* a     = (const float*)d_in[4];
  const float* biasW = (const float*)d_in[5];
  const float* selfW = (const float*)d_in[6];
  float* H = (float*)d_out;

  const size_t buf = (size_t)NBATCH * N_NODES * F;   // 2M floats = 8 MB
  float* Kb = (float*)d_ws;          // X @ att_W
  float* Vb = Kb + buf;              // X @ W
  float* Sf = Vb + buf;              // X @ self_W + bias

  const int blocks = NBATCH * WGS_PER_BATCH;         // 256 WGs, 8 waves each
  gat_phase0<<<blocks, 256, 0, stream>>>(X, W, attW, selfW, biasW, Kb, Vb, Sf);
  gat_phase1<<<blocks, 256, 0, stream>>>(a, Kb, Vb, Sf, H);
}